// BagModel_6536940225207
// MI455X (gfx1250) — compile-verified
//
#include <hip/hip_runtime.h>

typedef __attribute__((ext_vector_type(2))) float v2f;
typedef __attribute__((ext_vector_type(8))) float v8f;

// Problem dims fixed by the reference model (N, B taken at runtime).
#define D_DIM   128
#define H_DIM   256
#define F_DIM   64
#define MTILE   64

// LDS layouts (floats).
// Weights are stored "fragment-ready": for each even k, elements
// W[k][n], W[k+1][n] are adjacent so one ds_load_b64 fetches a B fragment.
// Row stride of a k-pair chosen with stride % 64 == 32 so lanes 16-31
// (k+2 row) hit the opposite half of the 64 LDS banks.
#define W1_STRIDE2 544   // 2*H_DIM + 32 ; 64 k-pairs  -> 34816 floats
#define W2_STRIDE2 160   // 2*F_DIM + 32 ; 128 k-pairs -> 20480 floats
#define X_STRIDE   132   // 64 x 132 (A-frag rows land on distinct banks)
#define H_STRIDE   260   // 64 x 260
#define F_STRIDE   68    // feats tile aliases the x tile region

__global__ void __launch_bounds__(256)
bag_zero_kernel(float* p, int n) {
    int i = blockIdx.x * blockDim.x + threadIdx.x;
    if (i < n) p[i] = 0.0f;
}

__global__ void __launch_bounds__(256)
bag_fused_kernel(const float* __restrict__ x,
                 const int*   __restrict__ ids,
                 const float* __restrict__ W1,
                 const float* __restrict__ b1,
                 const float* __restrict__ W2,
                 const float* __restrict__ b2,
                 float* __restrict__ sums,     // [B, F]
                 float* __restrict__ counts,   // [B]
                 int nRows, int numTiles)
{
    __shared__ float s_w1[(D_DIM / 2) * W1_STRIDE2];   // 136 KB
    __shared__ float s_w2[(H_DIM / 2) * W2_STRIDE2];   //  80 KB
    __shared__ float s_b1[H_DIM];
    __shared__ float s_b2[F_DIM];
    __shared__ float s_x [MTILE * X_STRIDE];           //  33 KB (aliased as feats)
    __shared__ float s_h [MTILE * H_STRIDE];           //  65 KB
    __shared__ int   s_bag[MTILE];

    const int tid   = threadIdx.x;
    const int w     = tid >> 5;         // wave id 0..7
    const int lane  = tid & 31;
    const int lhalf = lane >> 4;        // 0: lanes 0-15, 1: lanes 16-31
    const int lmod  = lane & 15;

    // ---- stage weights + biases into LDS once per persistent block ----
    // pairwise layout: s_w1[(k>>1)*S + 2n + (k&1)] = W1[k][n]
    for (int i = tid; i < D_DIM * H_DIM; i += 256) {
        int k = i >> 8, n = i & 255;
        s_w1[(k >> 1) * W1_STRIDE2 + 2 * n + (k & 1)] = W1[i];
    }
    for (int i = tid; i < H_DIM * F_DIM; i += 256) {
        int k = i >> 6, n = i & 63;
        s_w2[(k >> 1) * W2_STRIDE2 + 2 * n + (k & 1)] = W2[i];
    }
    s_b1[tid] = b1[tid];
    if (tid < F_DIM) s_b2[tid] = b2[tid];
    __syncthreads();

    float* fbuf = s_x;   // feats tile aliases the x tile region

    for (int tile = blockIdx.x; tile < numTiles; tile += gridDim.x) {
        const int row0 = tile * MTILE;

        // ---- load x tile (float4, coalesced) + bag ids ----
        for (int i4 = tid; i4 < MTILE * (D_DIM / 4); i4 += 256) {
            int r  = i4 >> 5;           // D_DIM/4 == 32 float4 per row
            int c4 = i4 & 31;
            int g  = row0 + r;
            float4 val = make_float4(0.f, 0.f, 0.f, 0.f);
            if (g < nRows)
                val = *(const float4*)(x + (size_t)g * D_DIM + c4 * 4);
            *(float4*)&s_x[r * X_STRIDE + c4 * 4] = val;
        }
        if (tid < MTILE) {
            int g = row0 + tid;
            s_bag[tid] = (g < nRows) ? ids[g] : -1;
        }
        __syncthreads();

        // ---- GEMM1: h = relu(x_tile @ W1 + b1), 64x256, K=128 ----
        // 64 output tiles (4 row x 16 col); wave w owns row-tile w/2,
        // col-tiles (w&1)*8 .. +7  -> A fragment shared across 8 tiles.
        {
            const int mr     = w >> 1;
            const int ncBase = (w & 1) * 8;
            const int arow   = mr * 16 + lmod;
            v8f acc[8];
            #pragma unroll
            for (int t = 0; t < 8; ++t) {
                float bv = s_b1[(ncBase + t) * 16 + lmod];
                #pragma unroll
                for (int v = 0; v < 8; ++v) acc[t][v] = bv;
            }
            for (int k0 = 0; k0 < D_DIM; k0 += 4) {
                const int kb    = k0 + (lhalf << 1);      // lanes>=16: k0+2
                const int kpair = kb >> 1;                // pair row in LDS
                v2f a = *(const v2f*)&s_x[arow * X_STRIDE + kb];
                #pragma unroll
                for (int t = 0; t < 8; ++t) {
                    const int col = (ncBase + t) * 16 + lmod;
                    v2f b = *(const v2f*)&s_w1[kpair * W1_STRIDE2 + 2 * col];
                    acc[t] = __builtin_amdgcn_wmma_f32_16x16x4_f32(
                        false, a, false, b, (short)0, acc[t], false, false);
                }
            }
            #pragma unroll
            for (int t = 0; t < 8; ++t) {
                const int hcol = (ncBase + t) * 16 + lmod;
                #pragma unroll
                for (int v = 0; v < 8; ++v) {
                    const int hrow = mr * 16 + v + (lhalf << 3);
                    float r = acc[t][v];
                    s_h[hrow * H_STRIDE + hcol] = (r > 0.0f) ? r : 0.0f;
                }
            }
        }
        __syncthreads();

        // ---- GEMM2: feats = h_tile @ W2 + b2, 64x64, K=256 ----
        // 16 output tiles (4x4); wave w owns row-tile w/2, col-tiles (w&1)*2, +1.
        {
            const int mr   = w >> 1;
            const int nc0  = (w & 1) * 2;
            const int col0 = nc0 * 16 + lmod;
            const int col1 = col0 + 16;
            const int arow = mr * 16 + lmod;
            v8f f0, f1;
            {
                float bv0 = s_b2[col0], bv1 = s_b2[col1];
                #pragma unroll
                for (int v = 0; v < 8; ++v) { f0[v] = bv0; f1[v] = bv1; }
            }
            for (int k0 = 0; k0 < H_DIM; k0 += 4) {
                const int kb    = k0 + (lhalf << 1);
                const int kpair = kb >> 1;
                v2f a  = *(const v2f*)&s_h[arow * H_STRIDE + kb];
                v2f bA = *(const v2f*)&s_w2[kpair * W2_STRIDE2 + 2 * col0];
                v2f bB = *(const v2f*)&s_w2[kpair * W2_STRIDE2 + 2 * col1];
                f0 = __builtin_amdgcn_wmma_f32_16x16x4_f32(
                    false, a, false, bA, (short)0, f0, false, false);
                f1 = __builtin_amdgcn_wmma_f32_16x16x4_f32(
                    false, a, false, bB, (short)0, f1, false, false);
            }
            #pragma unroll
            for (int v = 0; v < 8; ++v) {
                const int frow = mr * 16 + v + (lhalf << 3);
                fbuf[frow * F_STRIDE + col0] = f0[v];
                fbuf[frow * F_STRIDE + col1] = f1[v];
            }
        }
        __syncthreads();

        // ---- segment reduce: ids are sorted, run-length compress atomics ----
        {
            const int seg   = tid >> 6;     // 4 segments of 16 rows
            const int col   = tid & 63;
            const int rbase = seg * 16;
            if (row0 + rbase < nRows) {
                float acc = 0.0f, cnt = 0.0f;
                int cur = s_bag[rbase];
                for (int r = rbase; r < rbase + 16; ++r) {
                    if (row0 + r >= nRows) break;
                    int bg = s_bag[r];
                    if (bg != cur) {
                        atomicAdd(&sums[(size_t)cur * F_DIM + col], acc);
                        if (col == 0) atomicAdd(&counts[cur], cnt);
                        acc = 0.0f; cnt = 0.0f; cur = bg;
                    }
                    acc += fbuf[r * F_STRIDE + col];
                    cnt += 1.0f;
                }
                if (cnt > 0.0f) {
                    atomicAdd(&sums[(size_t)cur * F_DIM + col], acc);
                    if (col == 0) atomicAdd(&counts[cur], cnt);
                }
            }
        }
        __syncthreads();   // before next tile overwrites s_x / s_bag
    }
}

// out[b] = (sums[b,:] . Wa) / counts[b] + ba   -- one wave32 per bag
__global__ void __launch_bounds__(256)
bag_finalize_kernel(const float* __restrict__ sums,
                    const float* __restrict__ counts,
                    const float* __restrict__ Wa,
                    const float* __restrict__ ba,
                    float* __restrict__ out, int B)
{
    const int wavesPerBlock = blockDim.x >> 5;
    const int bag  = blockIdx.x * wavesPerBlock + (threadIdx.x >> 5);
    const int lane = threadIdx.x & 31;
    if (bag >= B) return;
    const float* row = sums + (size_t)bag * F_DIM;
    float v = row[lane] * Wa[lane] + row[lane + 32] * Wa[lane + 32];
    #pragma unroll
    for (int m = 16; m > 0; m >>= 1) v += __shfl_xor(v, m, 32);
    if (lane == 0) out[bag] = v / counts[bag] + ba[0];
}

extern "C" void kernel_launch(void* const* d_in, const int* in_sizes, int n_in,
                              void* d_out, int out_size, void* d_ws, size_t ws_size,
                              hipStream_t stream) {
    const float* x   = (const float*)d_in[0];
    const int*   ids = (const int*)  d_in[1];
    const float* W1  = (const float*)d_in[2];
    const float* b1  = (const float*)d_in[3];
    const float* W2  = (const float*)d_in[4];
    const float* b2  = (const float*)d_in[5];
    const float* Wa  = (const float*)d_in[6];
    const float* ba  = (const float*)d_in[7];
    float* out = (float*)d_out;

    const int N = in_sizes[1];          // number of instances
    const int B = out_size;             // number of bags

    float* sums   = (float*)d_ws;                 // B * F
    float* counts = sums + (size_t)B * F_DIM;     // B

    const int zn = B * F_DIM + B;
    bag_zero_kernel<<<(zn + 255) / 256, 256, 0, stream>>>(sums, zn);

    const int numTiles = (N + MTILE - 1) / MTILE;
    int grid = numTiles < 512 ? numTiles : 512;   // persistent blocks, ~315KB LDS -> 1/WGP
    bag_fused_kernel<<<grid, 256, 0, stream>>>(x, ids, W1, b1, W2, b2,
                                               sums, counts, N, numTiles);

    bag_finalize_kernel<<<(B + 7) / 8, 256, 0, stream>>>(sums, counts, Wa, ba, out, B);
}